// ExtractModel_28363964023179
// MI455X (gfx1250) — compile-verified
//
#include <hip/hip_runtime.h>
#include <hip/hip_bf16.h>
#include <float.h>
#include <limits.h>

typedef __attribute__((ext_vector_type(2))) float v2f;
typedef __attribute__((ext_vector_type(8))) float v8f;

#define LU 33
#define KU 29
#define BB 4
#define LL 128
#define VV 2000
#define MM 10
#define WW 10
#define MIN_LEN 4
#define INS_DEL_C 3.5f
#define NTHREADS 256
#define SUB_ROWS 36            // 33 rows padded to 36 (9 K-steps * 4) with zeros

__global__ __launch_bounds__(NTHREADS)
void extract_model_kernel(const float* __restrict__ unit_aligner,   // (33,29)
                          const float* __restrict__ ins_del_scale,  // (1,)
                          const int*   __restrict__ unit_id_seqs,   // (4,128)
                          const int*   __restrict__ vocab_unit_ids, // (2000,10)
                          const int*   __restrict__ vocab_lengths,  // (2000,)
                          float*       __restrict__ out)            // nll | score | vocab
{
    __shared__ float subLds[SUB_ROWS * 32];  // sub[u][k], zero-padded, row stride 32
    __shared__ float swinLds[16 * 32];       // sub_win[w][k], row stride 32
    __shared__ int   winLds[16];
    __shared__ float lseLds[LU];
    __shared__ float red_v[WW * NTHREADS];
    __shared__ int   red_i[WW * NTHREADS];
    __shared__ float fin_v[WW];
    __shared__ int   fin_i[WW];

    const int tid = threadIdx.x;
    const int blk = blockIdx.x;          // b*L + l
    const int b   = blk >> 7;
    const int l   = blk & (LL - 1);
    const float ins_del = ins_del_scale[0] * INS_DEL_C;

    // ---- Phase 0: per-row logsumexp of unit_aligner; window ids ----
    if (tid < LU) {
        const float* row = unit_aligner + tid * KU;
        float mx = row[0];
        #pragma unroll
        for (int k = 1; k < KU; ++k) mx = fmaxf(mx, row[k]);
        float s = 0.f;
        #pragma unroll
        for (int k = 0; k < KU; ++k) s += expf(row[k] - mx);
        lseLds[tid] = mx + logf(s);
    }
    if (tid < 16) {
        int p = l + tid; if (p > LL - 1) p = LL - 1;
        winLds[tid] = (tid < WW) ? unit_id_seqs[b * LL + p] : -1;  // -1: never matches
    }
    __syncthreads();

    // sub[u][k] = lse[u] - a[u][k]; rows 33..35 and cols 29..31 zeroed so the
    // WMMA B-fragment loads below are unconditional (no EXEC divergence).
    for (int idx = tid; idx < SUB_ROWS * 32; idx += NTHREADS) {
        const int u = idx >> 5, k = idx & 31;
        subLds[idx] = (u < LU && k < KU) ? (lseLds[u] - unit_aligner[u * KU + k]) : 0.f;
    }
    __syncthreads();

    // ---- Phase 1 (wave 0, EXEC all-ones throughout): sub_win = onehot(win) @ sub ----
    // Exact math: A is 0/1 f32, B is f32, V_WMMA_F32_16X16X4_F32 accumulates in f32.
    if (tid < 32) {
        const int lane = tid;
        const int n    = lane & 15;        // M for A frag, N for B/D frags
        const int half = lane >> 4;
        const int kb   = half * 2;         // K slot base for this half-wave
        const int wm   = winLds[n];        // one-hot row id for A (row m = n)
        v8f acc0 = {}; v8f acc1 = {};
        #pragma unroll
        for (int s = 0; s < 9; ++s) {      // K = 36 >= 33, zero-padded
            const int kk = s * 4 + kb;
            v2f a, b0, b1;
            a.x = (wm == kk)     ? 1.0f : 0.0f;
            a.y = (wm == kk + 1) ? 1.0f : 0.0f;
            b0.x = subLds[kk * 32 + n];            // rows < 36, cols < 32: in-bounds
            b0.y = subLds[(kk + 1) * 32 + n];
            b1.x = subLds[kk * 32 + 16 + n];
            b1.y = subLds[(kk + 1) * 32 + 16 + n];
            acc0 = __builtin_amdgcn_wmma_f32_16x16x4_f32(false, a, false, b0,
                                                         (short)0, acc0, false, false);
            acc1 = __builtin_amdgcn_wmma_f32_16x16x4_f32(false, a, false, b1,
                                                         (short)0, acc1, false, false);
        }
        #pragma unroll
        for (int r = 0; r < 8; ++r) {
            const int row = half * 8 + r;
            swinLds[row * 32 + n]      = acc0[r];
            swinLds[row * 32 + 16 + n] = acc1[r];
        }
    }
    __syncthreads();

    // ---- Phase 2: banded edit-distance DP, one vocab word per lane-iteration ----
    float minv[WW]; int mini[WW];
    #pragma unroll
    for (int w = 0; w < WW; ++w) { minv[w] = FLT_MAX; mini[w] = INT_MAX; }

    const size_t nll_blk_base = (size_t)blk * WW * VV;

    for (int v = tid; v < VV; v += NTHREADS) {
        if (v + NTHREADS < VV)
            __builtin_prefetch(vocab_unit_ids + (size_t)(v + NTHREADS) * MM, 0, 1);

        int ids[MM];
        #pragma unroll
        for (int i = 0; i < MM; ++i) ids[i] = vocab_unit_ids[v * MM + i];
        int vl = vocab_lengths[v];
        vl = (vl < 1) ? 1 : (vl > MM ? MM : vl);

        float f[WW + 1];
        #pragma unroll
        for (int j = 0; j <= WW; ++j) f[j] = (float)j * ins_del;

        #pragma unroll
        for (int i = 1; i <= MM; ++i) {
            const int cid = ids[i - 1];
            float diag = f[0];
            f[0] = (float)i * ins_del;
            #pragma unroll
            for (int j = 1; j <= WW; ++j) {
                const float up = f[j];
                const float c  = swinLds[(j - 1) * 32 + cid];
                f[j] = fminf(fminf(up, f[j - 1]) + ins_del, diag + c);
                diag = up;
            }
            if (i == vl) {
                const size_t base = nll_blk_base + (size_t)v;
                #pragma unroll
                for (int j = 1; j <= WW; ++j) {
                    const float val = f[j];
                    out[base + (size_t)(j - 1) * VV] = val;   // coalesced in v
                    if (val < minv[j - 1]) { minv[j - 1] = val; mini[j - 1] = v; }
                }
            }
        }
    }

    // ---- Phase 3: block reduction (min over V per w, first-occurrence ties) ----
    #pragma unroll
    for (int w = 0; w < WW; ++w) {
        red_v[w * NTHREADS + tid] = minv[w];
        red_i[w * NTHREADS + tid] = mini[w];
    }
    __syncthreads();
    if (tid < WW) {
        float bv = FLT_MAX; int bi = INT_MAX;
        for (int t = 0; t < NTHREADS; ++t) {
            const float val = red_v[tid * NTHREADS + t];
            const int   idx = red_i[tid * NTHREADS + t];
            if (val < bv || (val == bv && idx < bi)) { bv = val; bi = idx; }
        }
        fin_v[tid] = bv; fin_i[tid] = bi;
    }
    __syncthreads();
    if (tid == 0) {
        float bs = -FLT_MAX; int bj = 0; bool any = false;
        #pragma unroll
        for (int w = 0; w < WW; ++w) {
            const int span = w + 1;
            const bool viable = (l + span <= LL) && (span >= MIN_LEN);
            if (viable) {
                const float sc = (float)span - fin_v[w];
                if (!any || sc > bs) { bs = sc; bj = w; any = true; }
            }
        }
        const size_t score_off = (size_t)BB * LL * WW * VV;        // 10,240,000
        const size_t vocab_off = score_off + (size_t)BB * LL;      // +512
        out[score_off + blk] = any ? bs : -INFINITY;   // argmax of all -inf -> j=0
        out[vocab_off + blk] = (float)fin_i[bj];
    }
}

extern "C" void kernel_launch(void* const* d_in, const int* in_sizes, int n_in,
                              void* d_out, int out_size, void* d_ws, size_t ws_size,
                              hipStream_t stream) {
    const float* unit_aligner   = (const float*)d_in[0];
    const float* ins_del_scale  = (const float*)d_in[1];
    const int*   unit_id_seqs   = (const int*)d_in[2];
    const int*   vocab_unit_ids = (const int*)d_in[3];
    const int*   vocab_lengths  = (const int*)d_in[4];
    float*       out            = (float*)d_out;

    dim3 grid(BB * LL);      // 512 blocks, one per (b, l)
    dim3 block(NTHREADS);    // 8 wave32 per block
    extract_model_kernel<<<grid, block, 0, stream>>>(
        unit_aligner, ins_del_scale, unit_id_seqs, vocab_unit_ids, vocab_lengths, out);
}